// RGCN_Time_53214644798145
// MI455X (gfx1250) — compile-verified
//
#include <hip/hip_runtime.h>

typedef float v2f __attribute__((ext_vector_type(2)));
typedef float v8f __attribute__((ext_vector_type(8)));

#define DIM 256
#define KC 32
#define BM 64
#define SA_STRIDE 40   // floats; 160B rows -> 16B-aligned, padded vs bank conflicts
#define SW_STRIDE 34   // floats; W staged col-major [n][k], 8B-aligned rows

// ---------------------------------------------------------------------------
// W[r,i,o] = sum_b w_rel[r,b] * w_bases[b,i,o]
__global__ void rgcn_compose_weights(const float* __restrict__ w_rel,
                                     const float* __restrict__ w_bases,
                                     float* __restrict__ W,
                                     int R, int Bn, int IO) {
    int i = blockIdx.x * blockDim.x + threadIdx.x;
    if (i >= R * IO) return;
    int r = i / IO;
    int io = i - r * IO;
    float acc = 0.f;
    for (int b = 0; b < Bn; ++b)
        acc = fmaf(w_rel[r * Bn + b], w_bases[(size_t)b * IO + io], acc);
    W[i] = acc;
}

// ---------------------------------------------------------------------------
// Y[nRows,256] = X[nRows,256] @ W[256,256]   (fp32 WMMA 16x16x4)
// Block: 256 thr (8 waves). Block tile 64(M) x 256(N).
// Wave w: rows (w&3)*16 .. +16, cols (w>>2)*128 .. +128 (8 N-subtiles).
__global__ __launch_bounds__(256) void rgcn_gemm_xw(
    const float* __restrict__ X,
    const float* __restrict__ W,
    float* __restrict__ Y,
    int nRows)
{
    __shared__ float sA[BM * SA_STRIDE];   // 64 x 32 (+pad)
    __shared__ float sW[DIM * SW_STRIDE];  // 256 x 32 (+pad), col-major [n][k]

    const int t       = threadIdx.x;
    const int lane    = t & 31;
    const int wave    = t >> 5;
    const int rowGrp  = (wave & 3) * 16;
    const int colHalf = (wave >> 2) * 128;
    const int l16     = lane & 15;
    const int lhi     = lane >> 4;          // 0/1
    const int blockRow = blockIdx.x * BM;

    v8f acc[8] = {};

    for (int kc = 0; kc < DIM; kc += KC) {
        // --- stage X tile: 64 rows x 32 k via async global->LDS (ASYNCcnt);
        //     issued first so it overlaps the synchronous W staging below ---
        {
            int r0 = t >> 3;            // 0..31
            int c4 = (t & 7) * 4;       // 0..28
            #pragma unroll
            for (int h = 0; h < 2; ++h) {
                int r  = r0 + h * 32;
                int gr = blockRow + r;
                gr = gr < nRows ? gr : (nRows - 1);   // clamp (no divergence)
                const float* src = X + (size_t)gr * DIM + kc + c4;
#if defined(__HIP_DEVICE_COMPILE__)
                unsigned lds_off = (unsigned)(size_t)&sA[r * SA_STRIDE + c4];
                asm volatile("global_load_async_to_lds_b128 %0, %1, off"
                             :: "v"(lds_off),
                                "v"((unsigned long long)(size_t)src)
                             : "memory");
#else
                const float4 v = *(const float4*)src;
                float* dst = &sA[r * SA_STRIDE + c4];
                dst[0] = v.x; dst[1] = v.y; dst[2] = v.z; dst[3] = v.w;
#endif
                if (kc + KC < DIM)
                    __builtin_prefetch(src + KC, 0, 1);
            }
        }
        // --- stage W tile transposed: sW[n][k], n=0..255, k=kc..kc+31 ---
        {
            int n4  = (t & 63) * 4;     // 0..252
            int kr0 = t >> 6;           // 0..3
            #pragma unroll
            for (int h = 0; h < 8; ++h) {
                int kr = kr0 + h * 4;
                const float4 v = *(const float4*)(W + (size_t)(kc + kr) * DIM + n4);
                sW[(n4 + 0) * SW_STRIDE + kr] = v.x;
                sW[(n4 + 1) * SW_STRIDE + kr] = v.y;
                sW[(n4 + 2) * SW_STRIDE + kr] = v.z;
                sW[(n4 + 3) * SW_STRIDE + kr] = v.w;
            }
        }
#if defined(__HIP_DEVICE_COMPILE__)
        asm volatile("s_wait_asynccnt 0x0" ::: "memory");
#endif
        __syncthreads();

        // --- 8 k-steps of K=4 fp32 WMMA, 8 independent N-subtile accums ---
        #pragma unroll
        for (int k2 = 0; k2 < KC; k2 += 4) {
            const int kk = k2 + 2 * lhi;     // A/B frag: K = 2*(lane/16) + j
            v2f a = *(const v2f*)&sA[(rowGrp + l16) * SA_STRIDE + kk];
            #pragma unroll
            for (int s = 0; s < 8; ++s) {
                const int n = colHalf + s * 16 + l16;
                v2f b = *(const v2f*)&sW[n * SW_STRIDE + kk];
                acc[s] = __builtin_amdgcn_wmma_f32_16x16x4_f32(
                    false, a, false, b, (short)0, acc[s], false, false);
            }
        }
        __syncthreads();
    }

    // --- store: C/D layout -> row = v + 8*(lane/16), col = lane%16 ---
    #pragma unroll
    for (int v = 0; v < 8; ++v) {
        int row = blockRow + rowGrp + v + 8 * lhi;
        if (row < nRows) {
            float* yr = Y + (size_t)row * DIM + colHalf + l16;
            #pragma unroll
            for (int s = 0; s < 8; ++s)
                yr[s * 16] = acc[s][v];
        }
    }
}

// ---------------------------------------------------------------------------
// out[map(row_e)] += val_e * Y[col_e]   (64 lanes/edge, float4 per lane)
// Guaranteed-hardware non-returning f32 atomics via inline asm:
// 4 x global_atomic_add_f32 off one 64-bit address with inst offsets.
__global__ __launch_bounds__(256) void rgcn_scatter_edges(
    const float* __restrict__ Y,
    const int* __restrict__ rows,
    const int* __restrict__ cols,
    const float* __restrict__ vals,
    const int* __restrict__ idx,     // nullptr -> identity
    float* __restrict__ out,
    int nE)
{
    const int g = threadIdx.x >> 6;   // 0..3 : edge slot within block
    const int p = threadIdx.x & 63;   // float4 slice of the 256-wide row
    for (int e = blockIdx.x * 4 + g; e < nE; e += gridDim.x * 4) {
        int   r = rows[e];
        int   c = cols[e];
        float v = vals[e];
        int orow = idx ? idx[r] : r;
        const float4 y = *(const float4*)(Y + (size_t)c * DIM + p * 4);
        float* o = out + (size_t)orow * DIM + p * 4;
#if defined(__HIP_DEVICE_COMPILE__)
        asm volatile(
            "global_atomic_add_f32 %0, %1, off\n\t"
            "global_atomic_add_f32 %0, %2, off offset:4\n\t"
            "global_atomic_add_f32 %0, %3, off offset:8\n\t"
            "global_atomic_add_f32 %0, %4, off offset:12"
            :: "v"((unsigned long long)(size_t)o),
               "v"(v * y.x), "v"(v * y.y), "v"(v * y.z), "v"(v * y.w)
            : "memory");
#else
        __hip_atomic_fetch_add(o + 0, v * y.x, __ATOMIC_RELAXED, __HIP_MEMORY_SCOPE_AGENT);
        __hip_atomic_fetch_add(o + 1, v * y.y, __ATOMIC_RELAXED, __HIP_MEMORY_SCOPE_AGENT);
        __hip_atomic_fetch_add(o + 2, v * y.z, __ATOMIC_RELAXED, __HIP_MEMORY_SCOPE_AGENT);
        __hip_atomic_fetch_add(o + 3, v * y.w, __ATOMIC_RELAXED, __HIP_MEMORY_SCOPE_AGENT);
#endif
    }
}

// ---------------------------------------------------------------------------
extern "C" void kernel_launch(void* const* d_in, const int* in_sizes, int n_in,
                              void* d_out, int out_size, void* d_ws, size_t ws_size,
                              hipStream_t stream) {
    const float* input         = (const float*)d_in[0];
    const int*   adj_rows      = (const int*)  d_in[1];
    const int*   adj_cols      = (const int*)  d_in[2];
    const float* adj_vals      = (const float*)d_in[3];
    const float* emb_prev      = (const float*)d_in[4];
    const int*   adj_prev_rows = (const int*)  d_in[5];
    const int*   adj_prev_cols = (const int*)  d_in[6];
    const float* adj_prev_vals = (const float*)d_in[7];
    const int*   idx_prev_now  = (const int*)  d_in[8];
    const float* w_rel         = (const float*)d_in[9];
    const float* w_bases       = (const float*)d_in[10];
    const float* weight_prev   = (const float*)d_in[11];

    const int R  = 4;
    const int N  = in_sizes[0] / DIM;
    const int E  = in_sizes[1] / R;
    const int EP = in_sizes[5] / R;
    const int Bn = in_sizes[9] / R;
    const int IO = in_sizes[10] / Bn;          // D_IN * D_OUT = 65536

    float* Wc = (float*)d_ws;                  // R*IO floats (composed weights)
    float* Y  = Wc + (size_t)R * IO;           // N*256 floats (per-relation X@W)
    float* out = (float*)d_out;

    hipMemsetAsync(out, 0, (size_t)out_size * sizeof(float), stream);

    {
        int total = R * IO;
        rgcn_compose_weights<<<(total + 255) / 256, 256, 0, stream>>>(
            w_rel, w_bases, Wc, R, Bn, IO);
    }

    const int gemmGrid = (N + BM - 1) / BM;
    int scatGrid  = (E + 3) / 4;  if (scatGrid  > 16384) scatGrid  = 16384;
    int scatGridP = (EP + 3) / 4; if (scatGridP > 16384) scatGridP = 16384;

    for (int r = 0; r < R; ++r) {
        rgcn_gemm_xw<<<gemmGrid, 256, 0, stream>>>(input, Wc + (size_t)r * IO, Y, N);
        rgcn_scatter_edges<<<scatGrid, 256, 0, stream>>>(
            Y, adj_rows + (size_t)r * E, adj_cols + (size_t)r * E,
            adj_vals + (size_t)r * E, nullptr, out, E);
    }
    for (int r = 0; r < R; ++r) {
        rgcn_gemm_xw<<<gemmGrid, 256, 0, stream>>>(emb_prev, weight_prev + (size_t)r * IO, Y, N);
        rgcn_scatter_edges<<<scatGridP, 256, 0, stream>>>(
            Y, adj_prev_rows + (size_t)r * EP, adj_prev_cols + (size_t)r * EP,
            adj_prev_vals + (size_t)r * EP, idx_prev_now, out, EP);
    }
}